// GraphConvolution_6597069767349
// MI455X (gfx1250) — compile-verified
//
#include <hip/hip_runtime.h>

#define N_NODES 100000
#define N_EDGES 3200000
#define D 256          // D_IN == D_OUT == 256

typedef __attribute__((ext_vector_type(2))) float v2f;
typedef __attribute__((ext_vector_type(8))) float v8f;

// ---------------------------------------------------------------------------
// Kernel 1: support = x @ W   (FP32 WMMA 16x16x4)
//           out     = bias + (residual ? support : 0)
// grid.x = N_NODES/16 = 6250 blocks, 512 threads = 16 waves,
// wave w computes the 16x16 tile at cols [16w, 16w+16).
// ---------------------------------------------------------------------------
__global__ __launch_bounds__(512)
void gcn_gemm_wmma(const float* __restrict__ x,
                   const float* __restrict__ W,
                   const float* __restrict__ bias,
                   const int*   __restrict__ residual,
                   float* __restrict__ support,
                   float* __restrict__ out)
{
    const int lane = threadIdx.x & 31;
    const int wave = threadIdx.x >> 5;
    const int lr   = lane & 15;   // lane row/col within half-wave
    const int lh   = lane >> 4;   // which half of the wave
    const int r0   = blockIdx.x * 16;
    const int n0   = wave * 16;

    // A (16x4 f32): lanes 0-15 hold M=lane, K={k0,k0+1}; lanes 16-31 K={k0+2,k0+3}
    const float* xrow = x + (size_t)(r0 + lr) * D;
    // B (4x16 f32): row K striped across lanes within a VGPR, N = lr
    const float* wcol = W + n0 + lr;

    v8f acc = {};
    #pragma unroll 8
    for (int k0 = 0; k0 < D; k0 += 4) {
        const int ka = k0 + 2 * lh;
        v2f a;
        a.x = xrow[ka];
        a.y = xrow[ka + 1];
        v2f b;
        b.x = wcol[(size_t)ka * D];
        b.y = wcol[(size_t)(ka + 1) * D];
        // 8 args: (neg_a, A, neg_b, B, c_mod, C, reuse_a, reuse_b)
        acc = __builtin_amdgcn_wmma_f32_16x16x4_f32(
            false, a, false, b, (short)0, acc, false, false);
    }

    const int   res = *residual;
    const float bn  = bias[n0 + lr];

    // C/D layout: VGPR i -> M = i (lanes 0-15) or M = 8+i (lanes 16-31), N = lr
    #pragma unroll
    for (int i = 0; i < 8; ++i) {
        const int row  = r0 + lh * 8 + i;
        const size_t o = (size_t)row * D + (n0 + lr);
        const float s  = acc[i];
        support[o] = s;
        out[o]     = bn + (res ? s : 0.0f);
    }
}

// ---------------------------------------------------------------------------
// Kernel 2: COO scatter-add.  One wave per edge:
//   out[dst, :] += val * support[src, :]
// float4 gathers (global_load_b128) + global_atomic_add_f32 scatters.
// support/out are ~L2-resident (102.4 MB each vs 192 MB L2).
// ---------------------------------------------------------------------------
__global__ __launch_bounds__(256)
void gcn_scatter(const int*   __restrict__ edge_row,
                 const int*   __restrict__ edge_col,
                 const float* __restrict__ edge_val,
                 const float* __restrict__ support,
                 float* __restrict__ out)
{
    const int lane = threadIdx.x & 31;
    const int wave = threadIdx.x >> 5;
    const int e    = blockIdx.x * 8 + wave;
    if (e >= N_EDGES) return;

    const int   dst = edge_row[e];
    const int   src = edge_col[e];
    const float v   = edge_val[e];

    const float4* srow = (const float4*)(support + (size_t)src * D);
    float*        orow = out + (size_t)dst * D;

    #pragma unroll
    for (int h = 0; h < 2; ++h) {
        const int    q = h * 32 + lane;   // float4 index within the row (0..63)
        const float4 m = srow[q];
        const int    c = q * 4;
        atomicAdd(orow + c + 0, v * m.x);
        atomicAdd(orow + c + 1, v * m.y);
        atomicAdd(orow + c + 2, v * m.z);
        atomicAdd(orow + c + 3, v * m.w);
    }
}

// ---------------------------------------------------------------------------
extern "C" void kernel_launch(void* const* d_in, const int* in_sizes, int n_in,
                              void* d_out, int out_size, void* d_ws, size_t ws_size,
                              hipStream_t stream)
{
    const float* x        = (const float*)d_in[0];
    const int*   edge_row = (const int*)  d_in[1];
    const int*   edge_col = (const int*)  d_in[2];
    const float* edge_val = (const float*)d_in[3];
    const float* weight   = (const float*)d_in[4];
    const float* bias     = (const float*)d_in[5];
    const int*   residual = (const int*)  d_in[6];

    float* out     = (float*)d_out;
    float* support = (float*)d_ws;     // needs N_NODES * D * 4 = 102.4 MB

    // Phase 1: dense GEMM with f32 WMMA + fused bias/residual epilogue.
    gcn_gemm_wmma<<<N_NODES / 16, 512, 0, stream>>>(
        x, weight, bias, residual, support, out);

    // Phase 2: edge scatter (stream order provides the dependency).
    gcn_scatter<<<(N_EDGES + 7) / 8, 256, 0, stream>>>(
        edge_row, edge_col, edge_val, support, out);
}